// ScaledDotProductAttention_84628035600700
// MI455X (gfx1250) — compile-verified
//
#include <hip/hip_runtime.h>

// Fused scaled-dot-product attention for MI455X (gfx1250, wave32).
//   out  = softmax(mask ? -inf : QK^T/sqrt(512)) @ V      [128,1024,64] f32
//   attn = the softmax probabilities                       [128,1024,1024] f32
// d_out = out ++ attn (flat, in return order).
// Memory-bound on the 512MB attn write -> single fused pass, scores live in
// registers, WMMA f32 16x16x4 for both GEMMs.

#define BHN 128
#define SEQ 1024
#define DK  64
#define QT  16          // q rows per block
#define NWAVE 8
#define CT_PER_WAVE 8   // 16-wide column tiles per wave (8*16 = 128 cols/wave)

typedef __attribute__((ext_vector_type(2))) float v2f;
typedef __attribute__((ext_vector_type(8))) float v8f;

__device__ __forceinline__ v8f wmma4(v2f a, v2f b, v8f c) {
  // D = A(16x4 f32) * B(4x16 f32) + C(16x16 f32)
  return __builtin_amdgcn_wmma_f32_16x16x4_f32(
      /*neg_a=*/false, a, /*neg_b=*/false, b,
      /*c_mod=*/(short)0, c, /*reuse_a=*/false, /*reuse_b=*/false);
}

__launch_bounds__(256)
__global__ void attn_fused_kernel(const float* __restrict__ qp,
                                  const float* __restrict__ kp,
                                  const float* __restrict__ vp,
                                  const unsigned char* __restrict__ maskp,
                                  float* __restrict__ outp,
                                  float* __restrict__ attnp)
{
  __shared__ float ldsQ[QT * DK];           // 4 KB   Q tile
  __shared__ float ldsO[QT * DK];           // 4 KB   O accumulator
  __shared__ float stage[NWAVE][16 * 17];   // 8.5 KB per-wave P transpose staging
  __shared__ float redMax[NWAVE * QT];      // cross-wave row max
  __shared__ float redSum[NWAVE * QT];      // cross-wave row sum

  const int tid  = threadIdx.x;
  const int wave = tid >> 5;
  const int lane = tid & 31;
  const int half = lane >> 4;     // C/D layout: lanes 0-15 rows 0-7, 16-31 rows 8-15
  const int hl   = lane & 15;
  const int g2   = half * 2;      // A/B layout: lane half selects K pair {0,1} vs {2,3}

  const int qt = blockIdx.x;      // 0..63  (q tile)
  const int b  = blockIdx.y;      // 0..127 (batch*head)
  const int qrow0 = qt * QT;
  const size_t bS = (size_t)b * SEQ;

  // ---- stage Q tile, zero O accumulator (coalesced) ----
  for (int i = tid; i < QT * DK; i += 256) {
    ldsQ[i] = qp[(bS + qrow0 + (i >> 6)) * DK + (i & 63)];
    ldsO[i] = 0.0f;
  }
  __syncthreads();

  // ---- S = Q K^T / sqrt(512), masked; wave owns 128 columns, kept in VGPRs ----
  const float scale = 0.04419417382415922f;   // 1/sqrt(512)
  v8f p[CT_PER_WAVE];
#pragma unroll
  for (int ci = 0; ci < CT_PER_WAVE; ++ci) {
    const int ct   = wave * CT_PER_WAVE + ci;
    const int ncol = ct * 16 + hl;                 // K row / score column
    const float* krow = kp + (bS + ncol) * DK;
    if (ci + 1 < CT_PER_WAVE)
      __builtin_prefetch(krow + 16 * DK, 0, 3);    // global_prefetch_b8 next tile
    v8f acc = {};
#pragma unroll
    for (int d0 = 0; d0 < DK; d0 += 4) {
      v2f a, bb;
      a.x  = ldsQ[hl * DK + d0 + g2];
      a.y  = ldsQ[hl * DK + d0 + g2 + 1];
      bb.x = krow[d0 + g2];
      bb.y = krow[d0 + g2 + 1];
      acc = wmma4(a, bb, acc);
    }
    const unsigned char* mbase = maskp + (bS + qrow0) * SEQ + ncol;
#pragma unroll
    for (int r = 0; r < 8; ++r) {
      const int m = r + half * 8;
      p[ci][r] = mbase[(size_t)m * SEQ] ? -__builtin_inff() : acc[r] * scale;
    }
  }

  // ---- softmax: half-wave shfl reduce, then 8-wave LDS reduce ----
  float gmax[8];
#pragma unroll
  for (int r = 0; r < 8; ++r) {
    float mx = p[0][r];
#pragma unroll
    for (int ci = 1; ci < CT_PER_WAVE; ++ci) mx = fmaxf(mx, p[ci][r]);
#pragma unroll
    for (int sh = 1; sh < 16; sh <<= 1)            // stays within 16-lane half
      mx = fmaxf(mx, __shfl_xor(mx, sh, 32));
    gmax[r] = mx;
  }
  if (hl == 0) {
#pragma unroll
    for (int r = 0; r < 8; ++r) redMax[wave * QT + r + half * 8] = gmax[r];
  }
  __syncthreads();
#pragma unroll
  for (int r = 0; r < 8; ++r) {
    const int m = r + half * 8;
    float mx = redMax[m];
#pragma unroll
    for (int w = 1; w < NWAVE; ++w) mx = fmaxf(mx, redMax[w * QT + m]);
    gmax[r] = mx;
  }

  float rsum[8];
#pragma unroll
  for (int r = 0; r < 8; ++r) rsum[r] = 0.0f;
#pragma unroll
  for (int ci = 0; ci < CT_PER_WAVE; ++ci) {
#pragma unroll
    for (int r = 0; r < 8; ++r) {
      float e = __expf(p[ci][r] - gmax[r]);        // exp(-inf)=0 for masked
      p[ci][r] = e;
      rsum[r] += e;
    }
  }
#pragma unroll
  for (int r = 0; r < 8; ++r) {
    float s = rsum[r];
#pragma unroll
    for (int sh = 1; sh < 16; sh <<= 1)
      s += __shfl_xor(s, sh, 32);
    rsum[r] = s;
  }
  if (hl == 0) {
#pragma unroll
    for (int r = 0; r < 8; ++r) redSum[wave * QT + r + half * 8] = rsum[r];
  }
  __syncthreads();
  float ginv[8];
#pragma unroll
  for (int r = 0; r < 8; ++r) {
    const int m = r + half * 8;
    float s = redSum[m];
#pragma unroll
    for (int w = 1; w < NWAVE; ++w) s += redSum[w * QT + m];
    ginv[r] = 1.0f / s;
  }

  // ---- normalize, write attn, P·V partial (wave's own 128-column K range) ----
  v8f oacc[4] = {};
  float* st = stage[wave];
#pragma unroll
  for (int ci = 0; ci < CT_PER_WAVE; ++ci) {
    const int ct = wave * CT_PER_WAVE + ci;
    const int ncol = ct * 16 + hl;
#pragma unroll
    for (int r = 0; r < 8; ++r) {
      const int m = r + half * 8;
      float pv = p[ci][r] * ginv[r];
      attnp[(bS + qrow0 + m) * SEQ + ncol] = pv;   // 512MB stream, 64B/half-wave
      st[m * 17 + hl] = pv;                        // padded transpose staging
    }
    // Fence wave-internal LDS write->read (cross-lane) before A-fragment reads.
    asm volatile("s_wait_dscnt 0" ::: "memory");
#pragma unroll
    for (int n0 = 0; n0 < 16; n0 += 4) {
      v2f a;
      a.x = st[hl * 17 + n0 + g2];                 // P as 16x4 A fragment
      a.y = st[hl * 17 + n0 + g2 + 1];
      const int n = ct * 16 + n0 + g2;             // V row
      const float* vrow = vp + (bS + n) * DK + hl;
#pragma unroll
      for (int t = 0; t < 4; ++t) {                // 4 output 16-col tiles
        v2f bb;
        bb.x = vrow[t * 16];
        bb.y = vrow[t * 16 + DK];
        oacc[t] = wmma4(a, bb, oacc[t]);
      }
    }
  }

  // ---- combine the 8 wave partials via LDS float atomics, write O ----
#pragma unroll
  for (int t = 0; t < 4; ++t) {
#pragma unroll
    for (int r = 0; r < 8; ++r)
      atomicAdd(&ldsO[(r + half * 8) * DK + t * 16 + hl], oacc[t][r]);
  }
  __syncthreads();
  for (int i = tid; i < QT * DK; i += 256)
    outp[(bS + qrow0 + (i >> 6)) * DK + (i & 63)] = ldsO[i];
}

extern "C" void kernel_launch(void* const* d_in, const int* in_sizes, int n_in,
                              void* d_out, int out_size, void* d_ws, size_t ws_size,
                              hipStream_t stream) {
  (void)n_in; (void)out_size; (void)d_ws; (void)ws_size;
  const float* q = (const float*)d_in[0];
  const float* k = (const float*)d_in[1];
  const float* v = (const float*)d_in[2];
  const unsigned char* mask = (const unsigned char*)d_in[3];  // jnp bool = 1 byte
  float* out  = (float*)d_out;
  float* attn = out + (size_t)in_sizes[0];   // output elems = BH*S*DK, attn follows
  dim3 grid(SEQ / QT, BHN);
  attn_fused_kernel<<<grid, dim3(256), 0, stream>>>(q, k, v, mask, out, attn);
}